// SpatialAttention_54683523612761
// MI455X (gfx1250) — compile-verified
//
#include <hip/hip_runtime.h>
#include <math.h>

// Problem constants (hardcoded in reference)
#define B_  64
#define V_  207
#define VP  208      // 13 * 16, padded V for WMMA tiling
#define F_  2
#define T_  12
#define NT  13       // VP / 16 tiles per dim

typedef __attribute__((ext_vector_type(2))) float v2f;
typedef __attribute__((ext_vector_type(8))) float v8f;

// ---------------------------------------------------------------------------
// Prep 1: lhs[b,v,t] = sum_f (sum_t' x[b,v,f,t'] * w1[t']) * w2[f,t]
// Stored padded: (B, VP, T), rows v>=V_ zeroed (so WMMA A-frag padding is 0).
// ---------------------------------------------------------------------------
__global__ void k_prep_lhs(const float* __restrict__ x, const float* __restrict__ w1,
                           const float* __restrict__ w2, float* __restrict__ lhs) {
    int idx = blockIdx.x * blockDim.x + threadIdx.x;   // one thread per (b,v)
    if (idx >= B_ * VP) return;
    int b = idx / VP, v = idx % VP;
    float* o = lhs + (size_t)idx * T_;
    if (v < V_) {
        const float* xr = x + (size_t)(b * V_ + v) * F_ * T_;  // [f][t'] block
        float in0 = 0.f, in1 = 0.f;
#pragma unroll
        for (int t = 0; t < T_; ++t) {
            in0 += xr[t]        * w1[t];
            in1 += xr[T_ + t]   * w1[t];
        }
#pragma unroll
        for (int t = 0; t < T_; ++t)
            o[t] = in0 * w2[t] + in1 * w2[T_ + t];
    } else {
#pragma unroll
        for (int t = 0; t < T_; ++t) o[t] = 0.f;
    }
}

// ---------------------------------------------------------------------------
// Prep 2: rhs[b,t,u] = w3[0]*x2[0,b,t,u] + w3[1]*x2[1,b,t,u]
// where x2 = x viewed as (F,B,T,V) on the contiguous buffer.
// Stored padded: (B, T, VP), cols u>=V_ zeroed.
// ---------------------------------------------------------------------------
__global__ void k_prep_rhs(const float* __restrict__ x, const float* __restrict__ w3,
                           float* __restrict__ rhs) {
    int idx = blockIdx.x * blockDim.x + threadIdx.x;   // one thread per (b,t,u)
    if (idx >= B_ * T_ * VP) return;
    int b   = idx / (T_ * VP);
    int rem = idx % (T_ * VP);
    int t   = rem / VP, u = rem % VP;
    float val = 0.f;
    if (u < V_) {
        size_t base = (size_t)b * (T_ * V_) + (size_t)t * V_ + u;   // f = 0
        const size_t fstride = (size_t)B_ * T_ * V_;                // 158976
        val = w3[0] * x[base] + w3[1] * x[base + fstride];
    }
    rhs[idx] = val;   // idx == (b*T_ + t)*VP + u
}

// ---------------------------------------------------------------------------
// Prep 3: vsT[j,k] = v_s[k,j], padded (VP,VP) with zeros.
// ---------------------------------------------------------------------------
__global__ void k_prep_vst(const float* __restrict__ vs, float* __restrict__ vsT) {
    int idx = blockIdx.x * blockDim.x + threadIdx.x;
    if (idx >= VP * VP) return;
    int j = idx / VP, k = idx % VP;
    vsT[idx] = (j < V_ && k < V_) ? vs[k * V_ + j] : 0.f;
}

// ---------------------------------------------------------------------------
// GEMM 1 + sigmoid: sig[b,i,u] = sigmoid( sum_t lhs[b,i,t]*rhs[b,t,u] + b_s[i,u] )
// One wave32 per 16x16 tile; K=12 -> 3 x v_wmma_f32_16x16x4_f32.
// A frag (16x4 f32): lane<16 holds (M=lane, K=0..1); lane>=16 holds K=2..3.
// C/D (16x16 f32): vgpr r, lane -> (M = r + (lane>>4)*8, N = lane&15).
// ---------------------------------------------------------------------------
__global__ __launch_bounds__(32) void k_gemm_sig(const float* __restrict__ lhs,
                                                 const float* __restrict__ rhs,
                                                 const float* __restrict__ b_s,
                                                 float* __restrict__ sig) {
    const int lane = threadIdx.x;
    const int lr = lane & 15, hi = lane >> 4;
    const int i0 = blockIdx.x * 16, u0 = blockIdx.y * 16, b = blockIdx.z;

    const float* lrow  = lhs + ((size_t)b * VP + i0 + lr) * T_;     // A row for this lane
    const float* rbase = rhs + (size_t)b * T_ * VP + u0 + lr;       // B col for this lane

    v8f c = {};
#pragma unroll
    for (int s = 0; s < 3; ++s) {
        const int t = s * 4 + hi * 2;
        v2f a, bb;
        a.x  = lrow[t];
        a.y  = lrow[t + 1];
        bb.x = rbase[(size_t)t * VP];
        bb.y = rbase[(size_t)(t + 1) * VP];
        c = __builtin_amdgcn_wmma_f32_16x16x4_f32(false, a, false, bb,
                                                  (short)0, c, false, false);
    }

#pragma unroll
    for (int r = 0; r < 8; ++r) {
        const int i = i0 + r + hi * 8;
        const int u = u0 + lr;
        float val = 0.f;                       // zero padding -> K-dim of GEMM 2 is clean
        if (i < V_ && u < V_) {
            float p = c[r] + b_s[i * V_ + u];
            val = 1.f / (1.f + expf(-p));
        }
        sig[((size_t)b * VP + i) * VP + u] = val;
    }
}

// ---------------------------------------------------------------------------
// GEMM 2: S[b,i,k] = sum_j sig[b,i,j] * vsT[j,k]   (== sum_j v_s[k,j]*sig[b,i,j])
// One wave32 per 16x16 tile; K=208 -> 52 x v_wmma_f32_16x16x4_f32.
// Writes unpadded S straight into d_out.
// ---------------------------------------------------------------------------
__global__ __launch_bounds__(32) void k_gemm_s(const float* __restrict__ sig,
                                               const float* __restrict__ vsT,
                                               float* __restrict__ out) {
    const int lane = threadIdx.x;
    const int lr = lane & 15, hi = lane >> 4;
    const int i0 = blockIdx.x * 16, k0 = blockIdx.y * 16, b = blockIdx.z;

    const float* arow = sig + ((size_t)b * VP + i0 + lr) * VP;
    const float* bcol = vsT + k0 + lr;

    v8f c = {};
#pragma unroll 4
    for (int jt = 0; jt < VP; jt += 4) {
        const int j = jt + hi * 2;
        v2f a, bb;
        a.x  = arow[j];
        a.y  = arow[j + 1];
        bb.x = bcol[(size_t)j * VP];
        bb.y = bcol[(size_t)(j + 1) * VP];
        c = __builtin_amdgcn_wmma_f32_16x16x4_f32(false, a, false, bb,
                                                  (short)0, c, false, false);
    }

#pragma unroll
    for (int r = 0; r < 8; ++r) {
        const int i = i0 + r + hi * 8;
        const int k = k0 + lr;
        if (i < V_ && k < V_)
            out[((size_t)b * V_ + i) * V_ + k] = c[r];
    }
}

// ---------------------------------------------------------------------------
// Softmax over axis=1 (rows i), in place on d_out. Each thread owns one
// (b, k) column -> deterministic, no races. Stable max-subtract form.
// ---------------------------------------------------------------------------
__global__ __launch_bounds__(256) void k_softmax(float* __restrict__ out) {
    const int b = blockIdx.x;
    const int k = threadIdx.x;
    if (k >= V_) return;
    float* col = out + (size_t)b * V_ * V_ + k;

    float m = -INFINITY;
    for (int i = 0; i < V_; ++i)
        m = fmaxf(m, col[(size_t)i * V_]);

    float s = 0.f;
    for (int i = 0; i < V_; ++i) {
        float e = expf(col[(size_t)i * V_] - m);
        col[(size_t)i * V_] = e;
        s += e;
    }

    const float inv = 1.f / s;
    for (int i = 0; i < V_; ++i)
        col[(size_t)i * V_] *= inv;
}

// ---------------------------------------------------------------------------
extern "C" void kernel_launch(void* const* d_in, const int* in_sizes, int n_in,
                              void* d_out, int out_size, void* d_ws, size_t ws_size,
                              hipStream_t stream) {
    (void)in_sizes; (void)n_in; (void)out_size; (void)ws_size;

    const float* x  = (const float*)d_in[0];   // (B,V,F,T)
    const float* w1 = (const float*)d_in[1];   // (T,)
    const float* w2 = (const float*)d_in[2];   // (F,T)
    const float* w3 = (const float*)d_in[3];   // (F,)
    const float* bs = (const float*)d_in[4];   // (1,V,V)
    const float* vs = (const float*)d_in[5];   // (V,V)
    float* out = (float*)d_out;                // (B,V,V)

    // Workspace carve-up (~12.5 MB total, all fully rewritten every call)
    float* lhs = (float*)d_ws;                           // B*VP*T
    float* rhs = lhs + (size_t)B_ * VP * T_;             // B*T*VP
    float* vsT = rhs + (size_t)B_ * T_ * VP;             // VP*VP
    float* sig = vsT + (size_t)VP * VP;                  // B*VP*VP

    k_prep_lhs<<<(B_ * VP + 255) / 256, 256, 0, stream>>>(x, w1, w2, lhs);
    k_prep_rhs<<<(B_ * T_ * VP + 255) / 256, 256, 0, stream>>>(x, w3, rhs);
    k_prep_vst<<<(VP * VP + 255) / 256, 256, 0, stream>>>(vs, vsT);

    dim3 tiles(NT, NT, B_);
    k_gemm_sig<<<tiles, 32, 0, stream>>>(lhs, rhs, bs, sig);
    k_gemm_s  <<<tiles, 32, 0, stream>>>(sig, vsT, out);

    k_softmax<<<B_, 256, 0, stream>>>(out);
}